// DynamicInteraction_Layer_28991029248376
// MI455X (gfx1250) — compile-verified
//
#include <hip/hip_runtime.h>

typedef __attribute__((ext_vector_type(2))) float v2f;
typedef __attribute__((ext_vector_type(4))) float v4f;
typedef __attribute__((ext_vector_type(8))) float v8f;

#define B_  128
#define N_  64
#define D_  1024
#define P_  4
#define C_  4
#define ND  (N_ * D_)          // 65536 flattened n*d per batch
#define THRESH 0.7f
#define EPS_ 1e-8f

// ---------------------------------------------------------------------------
// Kernel 1: per (b,p) — normalize path probs, emit all_path_prob output,
// and build the gate-folded weight matrix W'[b][p][c] in workspace:
//   W'[p][c] = prob[p][c]/(sum+eps) + (c==0 && sum<0.7 ? 1 : 0)
// Folds the skip connection (adds emb0) exactly into the GEMM weights.
// ---------------------------------------------------------------------------
__global__ void di_weights_kernel(const float* __restrict__ path_prob,
                                  float* __restrict__ out_pp,   // d_out + B*P*ND
                                  float* __restrict__ wprime)   // ws: B*P*C floats
{
    int i = blockIdx.x * blockDim.x + threadIdx.x;   // over B*P
    if (i >= B_ * P_) return;
    v4f pp = *(const v4f*)(path_prob + i * 4);
    float s   = pp.x + pp.y + pp.z + pp.w;
    float inv = 1.0f / (s + EPS_);
    v4f npp = pp * inv;
    *(v4f*)(out_pp + i * 4) = npp;                   // all_path_prob output
    v4f w = npp;
    if (s < THRESH) w.x += 1.0f;                     // gate folded into c==0
    *(v4f*)(wprime + i * 4) = w;
}

// ---------------------------------------------------------------------------
// Kernel 2: streaming aggregation via V_WMMA_F32_16X16X4_F32.
// Per wave: one batch b, 256 consecutive x (16 WMMA tiles of 16).
//   Phase 1: issue all 32 global_load_b32 (B operands) -> deep MLP of loads
//   Phase 2: per tile-pair, 2 WMMAs; merge pair rows with ds_bpermute so all
//            32 lanes store 32 consecutive floats (full-wave 128B stores,
//            no exec-mask branching).
// Layouts (ISA 7.12.2, 32-bit):
//   A 16x4 : lane<16 {W'[m][0],W'[m][1]}, lane>=16 {W'[m][2],W'[m][3]}, m>=4 -> 0
//   B 4x16 : lane<16 {emb0[x0+L], emb1[x0+L]}, lane>=16 {emb2[..], emb3[..]}
//   D      : row p (p<4) = VGPR p, lanes 0..15
// ---------------------------------------------------------------------------
__global__ __launch_bounds__(256)
void di_aggr_wmma_kernel(const float* __restrict__ e0,
                         const float* __restrict__ e1,
                         const float* __restrict__ e2,
                         const float* __restrict__ e3,
                         const float* __restrict__ wprime,
                         float* __restrict__ out)
{
    const int lane = threadIdx.x & 31;
    const int wave = threadIdx.x >> 5;

    // 32 blocks per batch, 8 waves per block, 256 x per wave
    const int b     = blockIdx.x >> 5;
    const int xBase = ((blockIdx.x & 31) << 11) + (wave << 8);

    const int half = lane >> 4;      // 0: K=0,1 (emb0/emb1)   1: K=2,3 (emb2/emb3)
    const int l15  = lane & 15;

    // ---- A operand: gate-folded weights, rows m>=4 are zero ----
    v2f a = {0.0f, 0.0f};
    const int m  = l15;
    const int c0 = half ? 2 : 0;
    if (m < P_) {
        a = *(const v2f*)(wprime + (m * C_ + c0) + b * (P_ * C_));
    }

    // ---- Phase 1: prefetch all 16 B-operand tiles (32 loads in flight) ----
    const float* __restrict__ pX = half ? e2 : e0;   // B VGPR0: K = c0
    const float* __restrict__ pY = half ? e3 : e1;   // B VGPR1: K = c0+1
    const size_t ebase = (size_t)b * ND + (size_t)xBase + (size_t)l15;

    v2f bm[16];
    #pragma unroll
    for (int u = 0; u < 16; ++u) {
        bm[u].x = pX[ebase + u * 16];
        bm[u].y = pY[ebase + u * 16];
    }

    // ---- Phase 2: WMMA pairs + full-wave merged stores ----
    float* __restrict__ outp =
        out + (size_t)b * (size_t)(P_ * ND) + (size_t)xBase + (size_t)lane;

    const int permIdx = (lane & 15) * 4;   // lanes 16..31 gather src[lane-16]

    #pragma unroll
    for (int t = 0; t < 8; ++t) {
        v8f z0 = {};
        v8f d0 = __builtin_amdgcn_wmma_f32_16x16x4_f32(
            false, a, false, bm[2 * t + 0], (short)0, z0, false, false);
        v8f z1 = {};
        v8f d1 = __builtin_amdgcn_wmma_f32_16x16x4_f32(
            false, a, false, bm[2 * t + 1], (short)0, z1, false, false);

        const int xo = t * 32;
        #pragma unroll
        for (int p = 0; p < P_; ++p) {
            // move tile-1 row p (lanes 0..15) into lanes 16..31
            int hi = __builtin_amdgcn_ds_bpermute(permIdx,
                                                  __float_as_int(d1[p]));
            float v = half ? __int_as_float(hi) : d0[p];
            outp[xo + p * ND] = v;   // 32 consecutive floats, all lanes active
        }
    }
}

// ---------------------------------------------------------------------------
extern "C" void kernel_launch(void* const* d_in, const int* in_sizes, int n_in,
                              void* d_out, int out_size, void* d_ws, size_t ws_size,
                              hipStream_t stream) {
    const float* e0 = (const float*)d_in[0];
    const float* e1 = (const float*)d_in[1];
    const float* e2 = (const float*)d_in[2];
    const float* e3 = (const float*)d_in[3];
    const float* pp = (const float*)d_in[4];

    float* out    = (float*)d_out;
    float* out_pp = out + (size_t)B_ * P_ * ND;   // all_path_prob tail
    float* wprime = (float*)d_ws;                 // B*P*C gate-folded weights

    // Stage 1: weights + all_path_prob  (512 work items)
    di_weights_kernel<<<2, 256, 0, stream>>>(pp, out_pp, wprime);

    // Stage 2: WMMA streaming aggregation
    // grid = B * (ND / 2048) = 128 * 32 = 4096 blocks of 8 waves
    di_aggr_wmma_kernel<<<B_ * (ND / 2048), 256, 0, stream>>>(
        e0, e1, e2, e3, wprime, out);
}